// CHARM_40200893891073
// MI455X (gfx1250) — compile-verified
//
#include <hip/hip_runtime.h>
#include <hip/hip_bf16.h>

typedef __attribute__((ext_vector_type(2))) float v2f;
typedef __attribute__((ext_vector_type(8))) float v8f;

#define CH 64
#define DD 256

// ---------------------------------------------------------------------------
// Kernel 1: one workgroup (8 wave32s) computes remap[64][256] into workspace.
//   rowsum = W * ones       (waves 0-3, V_WMMA_F32_16X16X4_F32 accumulation)
//   colsum = ones^T * W     (waves 4-7)
//   remap  = exp(rowsum (x) colsum) / 256   (outer product via WMMA, K=1)
// ---------------------------------------------------------------------------
__global__ __launch_bounds__(256) void CHARM_remap_kernel(
    const float* __restrict__ W, float* __restrict__ remap) {
  __shared__ float s_rowsum[CH];
  __shared__ float s_colsum[DD];

  const int tid   = threadIdx.x;
  const int wave  = tid >> 5;    // 8 waves (wave32)
  const int lane  = tid & 31;
  const int lhalf = lane & 15;   // N (or M) position within half-wave
  const int khi   = lane >> 4;   // 0: lanes 0-15 (K=0,1), 1: lanes 16-31 (K=2,3)

  if (wave < 4) {
    // ---- rowsum for rows [wave*16, wave*16+16): C += A(Wtile 16x4) * B(ones 4x16)
    const int rowbase = wave * 16;
    v2f bones; bones.x = 1.0f; bones.y = 1.0f;
    v8f c = {};
    for (int k0 = 0; k0 < DD; k0 += 4) {
      // A layout: lane<16 holds A[M=lane][K=k0+0..1], lane>=16 holds K=k0+2..3
      const float* wrow = W + (size_t)(rowbase + lhalf) * DD + k0 + khi * 2;
      v2f a; a.x = wrow[0]; a.y = wrow[1];
      c = __builtin_amdgcn_wmma_f32_16x16x4_f32(
          false, a, false, bones, (short)0, c, false, false);
    }
    // C[m][n] == rowsum[m] for all n.  VGPR r: M=r (lanes0-15) / M=r+8 (lanes16-31)
    if (lhalf == 0) {
#pragma unroll
      for (int r = 0; r < 8; ++r) s_rowsum[rowbase + khi * 8 + r] = c[r];
    }
  } else {
    // ---- colsum for 4 column-tiles per wave: C += A(ones 16x4) * B(Wtile 4x16)
    const int ntile0 = (wave - 4) * 4;
    v2f aones; aones.x = 1.0f; aones.y = 1.0f;
    for (int t = 0; t < 4; ++t) {
      const int colbase = (ntile0 + t) * 16;
      v8f c = {};
      for (int k0 = 0; k0 < CH; k0 += 4) {
        // B layout: VGPR v holds row K=v (lanes0-15) and K=v+2 (lanes16-31)
        v2f b;
        b.x = W[(size_t)(k0 + 2 * khi + 0) * DD + colbase + lhalf];
        b.y = W[(size_t)(k0 + 2 * khi + 1) * DD + colbase + lhalf];
        c = __builtin_amdgcn_wmma_f32_16x16x4_f32(
            false, aones, false, b, (short)0, c, false, false);
      }
      // C[m][n] == colsum[colbase+n] for all m; take M=0 row (VGPR0, lanes 0-15)
      if (khi == 0) s_colsum[colbase + lhalf] = c[0];
    }
  }
  __syncthreads();

  // ---- outer product tiles (4 rowtiles x 16 ntiles = 64), 8 per wave, one WMMA each
  const float inv_d = 1.0f / (float)DD;
  for (int t = wave; t < 64; t += 8) {
    const int rt = t >> 4;   // 0..3
    const int nt = t & 15;   // 0..15
    // A: column K=0 = rowsum chunk, K=1..3 = 0
    v2f a; a.x = (khi == 0) ? s_rowsum[rt * 16 + lhalf] : 0.0f; a.y = 0.0f;
    // B: row K=0 = colsum chunk, K=1..3 = 0
    v2f b; b.x = (khi == 0) ? s_colsum[nt * 16 + lhalf] : 0.0f; b.y = 0.0f;
    v8f c = {};
    c = __builtin_amdgcn_wmma_f32_16x16x4_f32(
        false, a, false, b, (short)0, c, false, false);
#pragma unroll
    for (int r = 0; r < 8; ++r) {
      const int m = rt * 16 + khi * 8 + r;
      const int n = nt * 16 + lhalf;
      remap[m * DD + n] = __expf(c[r]) * inv_d;
    }
  }
}

// ---------------------------------------------------------------------------
// Kernel 2: broadcast remap (64 KB, cache-resident) to out[512][64][256].
// Pure store-bandwidth kernel: b128 loads (L2 hits) + b128 stores (32 MB).
// ---------------------------------------------------------------------------
__global__ __launch_bounds__(256) void CHARM_bcast_kernel(
    const float4* __restrict__ remap4, float4* __restrict__ out4,
    unsigned total4) {
  const unsigned stride = gridDim.x * blockDim.x;
  for (unsigned i = blockIdx.x * blockDim.x + threadIdx.x; i < total4;
       i += stride) {
    out4[i] = remap4[i & 4095u];  // 16384 floats = 4096 float4 per batch slice
  }
}

extern "C" void kernel_launch(void* const* d_in, const int* in_sizes, int n_in,
                              void* d_out, int out_size, void* d_ws,
                              size_t ws_size, hipStream_t stream) {
  // inputs: [0]=x (unused), [1]=conv_w (unused), [2]=conv_b (unused),
  //         [3]=emb_weight [64,256] f32
  const float* emb = (const float*)d_in[3];
  float* remap = (float*)d_ws;          // 64*256*4 = 64 KB scratch
  float* out = (float*)d_out;           // 512*64*256 f32

  CHARM_remap_kernel<<<1, 256, 0, stream>>>(emb, remap);

  const unsigned total4 = (unsigned)(out_size / 4);  // 2,097,152 float4
  CHARM_bcast_kernel<<<2048, 256, 0, stream>>>(
      (const float4*)remap, (float4*)out, total4);
}